// BatchAllTripletLoss_54125177864681
// MI455X (gfx1250) — compile-verified
//
#include <hip/hip_runtime.h>
#include <hip/hip_bf16.h>

typedef float v2f __attribute__((ext_vector_type(2)));
typedef float v8f __attribute__((ext_vector_type(8)));
typedef unsigned uint32x4 __attribute__((ext_vector_type(4)));
typedef int      int32x8  __attribute__((ext_vector_type(8)));
typedef int      int32x4  __attribute__((ext_vector_type(4)));

#define TWO_N  512   // 2N rows in batch
#define HALF_N 256   // N
#define DIMS   512   // feature dim
#define EPS_REL 1e-5f
#define EPS_CLAMP 1e-7f

// LDS A-panel row stride in dwords: 512 data + 4 pad dwords (2 dwords inserted
// by TDM after each 256-dword interval) -> banks lo*516 % 64 = lo*4, conflict-free.
#define LDS_ROW_STRIDE 516

__device__ __forceinline__ const float* row_ptr(const float* __restrict__ h1,
                                                const float* __restrict__ h2, int r) {
  return (r < HALF_N) ? (h1 + (size_t)r * DIMS) : (h2 + (size_t)(r - HALF_N) * DIMS);
}

// ---------------------------------------------------------------------------
// G = B * B^T.  Block = 8 waves computing a 16x128 strip: row panel row0..+15,
// columns blockIdx.x*128 + wave*16.  The 16x512 fp32 A panel (32 KB) is staged
// into LDS once per block by the Tensor Data Mover (tensor_load_to_lds), with
// TDM padding giving a 516-dword LDS row stride (bank-conflict-free b64 reads,
// 8B-aligned).  Each wave then runs 128 V_WMMA_F32_16X16X4_F32 accumulations,
// A fragments from LDS (ds_load_b64), B fragments from global (L2-resident).
// ---------------------------------------------------------------------------
__global__ void gram_wmma_tdm_kernel(const float* __restrict__ h1,
                                     const float* __restrict__ h2,
                                     float* __restrict__ G) {
  __shared__ float smem[16 * LDS_ROW_STRIDE + 8];   // ~33 KB

  const int lane = threadIdx.x & 31;
  const int wave = threadIdx.x >> 5;    // 0..7
  const int hi   = lane >> 4;           // 0/1 -> K offset 0/2
  const int lo   = lane & 15;
  const int row0 = blockIdx.y * 16;
  const int col0 = blockIdx.x * 128 + wave * 16;

  // ---- TDM: DMA the 16x512 A panel into LDS (wave 0 only; TENSORcnt is
  // per-wave, so the issuing wave waits, then the block barrier publishes).
  if (threadIdx.x < 32) {
    const float* apanel = row_ptr(h1, h2, row0);    // 16-aligned => single source
    unsigned long long ga = (unsigned long long)(size_t)apanel;
    unsigned lds_off = (unsigned)(size_t)(void*)smem;   // low 32 bits = LDS byte addr

    uint32x4 g0;
    g0.x = 1u;                                         // count=1 valid descriptor
    g0.y = lds_off;                                    // lds_addr [63:32]
    g0.z = (unsigned)(ga & 0xFFFFFFFFull);             // global_addr [95:64]->[31:0]
    g0.w = (unsigned)((ga >> 32) & 0x01FFFFFFull)      // global_addr [56:32]
         | (2u << 30);                                 // type = 2 ("image")

    int32x8 g1;
    g1[0] = (int)((2u << 16)      // data_size = 4 bytes
                | (1u << 20)      // pad_enable
                | (7u << 22)      // pad_interval: 256 dwords
                | (1u << 25));    // pad_amount: 2 dwords
    g1[1] = (int)(512u << 16);    // tensor_dim0 = 512  (bits 79:48)
    g1[2] = (int)(16u  << 16);    // tensor_dim1 = 16   (bits 111:80)
    g1[3] = (int)(512u << 16);    // tile_dim0  = 512   (bits 127:112)
    g1[4] = 16;                   // tile_dim1  = 16    (bits 143:128)
    g1[5] = 512;                  // tensor_dim0_stride = 512 (bits 207:160)
    g1[6] = 0;
    g1[7] = 0;

    int32x4 z4 = {0, 0, 0, 0};            // groups 2/3 unused (2D tensor)
    int32x8 z8 = {0, 0, 0, 0, 0, 0, 0, 0};
    __builtin_amdgcn_tensor_load_to_lds(g0, g1, z4, z4, z8, 0);
    __builtin_amdgcn_s_wait_tensorcnt(0);
  }
  __syncthreads();

  const float* arow = smem + lo * LDS_ROW_STRIDE;         // LDS, padded rows
  const float* brow = row_ptr(h1, h2, col0 + lo);         // global

  v8f c = {0.f, 0.f, 0.f, 0.f, 0.f, 0.f, 0.f, 0.f};

  // K in [0,256): no mid-row pad offset yet.
#pragma unroll 4
  for (int k = 0; k < 256; k += 4) {
    v2f a = *(const v2f*)(arow + k + 2 * hi);
    v2f b = *(const v2f*)(brow + k + 2 * hi);
    c = __builtin_amdgcn_wmma_f32_16x16x4_f32(false, a, false, b,
                                              (short)0, c, false, false);
  }
  // K in [256,512): skip the 2-dword mid-row pad.
#pragma unroll 4
  for (int k = 256; k < 512; k += 4) {
    v2f a = *(const v2f*)(arow + 2 + k + 2 * hi);
    v2f b = *(const v2f*)(brow + k + 2 * hi);
    c = __builtin_amdgcn_wmma_f32_16x16x4_f32(false, a, false, b,
                                              (short)0, c, false, false);
  }

  // C/D layout: VGPR i -> M = 8*hi + i, N = lo
#pragma unroll
  for (int i = 0; i < 8; ++i) {
    G[(size_t)(row0 + hi * 8 + i) * TWO_N + col0 + lo] = c[i];
  }
}

// ---------------------------------------------------------------------------
// acc[0] = sum of triplet values > eps (float)
// acc[1] = count of values > eps      (u32)
// acc[2] = count of "bad" (relu(v) >= eps) entries (u32)
// ---------------------------------------------------------------------------
__global__ void zero_acc_kernel(float* __restrict__ acc) {
  if (threadIdx.x < 4) ((unsigned*)acc)[threadIdx.x] = 0u;
}

__device__ __forceinline__ float dist_from_gram(float n2a, float n2b, float g) {
  float sq = fmaxf(n2a + n2b - 2.0f * g, 0.0f);
  float d  = (sq > 0.0f) ? sqrtf(sq) : 0.0f;   // torch.norm: 0 at 0
  return fmaxf(d, EPS_CLAMP);                  // clamp(min=1e-7)
}

// One block per anchor a; thread t handles p=t and p=t+256 (its partner).
__global__ void triplet_stats_kernel(const float* __restrict__ G,
                                     float* __restrict__ acc) {
  const int a = blockIdx.x;
  const int t = threadIdx.x;          // 0..255
  const int p = t;
  const int q = t + HALF_N;           // partner(p)

  const float n2a = G[(size_t)a * (TWO_N + 1)];
  const float n2p = G[(size_t)p * (TWO_N + 1)];
  const float n2q = G[(size_t)q * (TWO_N + 1)];

  const float dp = dist_from_gram(n2a, n2p, G[(size_t)a * TWO_N + p]);
  const float dq = dist_from_gram(n2a, n2q, G[(size_t)a * TWO_N + q]);

  const float v1 = dp - dq + 1.0f;    // triplet(a, p, partner(p))
  const float v2 = dq - dp + 1.0f;    // triplet(a, partner(p), p)

  float s = 0.0f; unsigned c = 0, b = 0;
  if (v1 > EPS_REL) { s += v1; ++c; }
  if (v2 > EPS_REL) { s += v2; ++c; }
  if (v1 >= EPS_REL) ++b;             // !(relu(v) < eps)  <=>  v >= eps
  if (v2 >= EPS_REL) ++b;

  __shared__ float    ssum[256];
  __shared__ unsigned scnt[256];
  __shared__ unsigned sbad[256];
  ssum[t] = s; scnt[t] = c; sbad[t] = b;
  __syncthreads();
#pragma unroll
  for (int off = 128; off > 0; off >>= 1) {
    if (t < off) {
      ssum[t] += ssum[t + off];
      scnt[t] += scnt[t + off];
      sbad[t] += sbad[t + off];
    }
    __syncthreads();
  }
  if (t == 0) {
    atomicAdd(&acc[0], ssum[0]);
    atomicAdd((unsigned*)acc + 1, scnt[0]);
    atomicAdd((unsigned*)acc + 2, sbad[0]);
  }
}

// ---------------------------------------------------------------------------
// out[0]=loss, out[1]=mean(differences)=0 exactly, out[2]=good, out[3]=bad,
// out[4]=sqrt(mean_norm_sq)
// ---------------------------------------------------------------------------
__global__ void finalize_kernel(const float* __restrict__ G,
                                const float* __restrict__ acc,
                                float* __restrict__ out) {
  __shared__ float sd[TWO_N];
  const int t = threadIdx.x;          // 0..511
  sd[t] = G[(size_t)t * (TWO_N + 1)]; // diag = squared row norms
  __syncthreads();
#pragma unroll
  for (int off = 256; off > 0; off >>= 1) {
    if (t < off) sd[t] += sd[t + off];
    __syncthreads();
  }
  if (t == 0) {
    const float mean_norm_sq = sd[0] / (float)TWO_N;
    const float    sum_rel = acc[0];
    const unsigned cnt     = ((const unsigned*)acc)[1];
    const unsigned bad     = ((const unsigned*)acc)[2];
    const float mean_rel = sum_rel / (float)cnt;

    const long long total = 134217728ll;            // 512^3
    out[0] = mean_rel + 1e-4f * mean_norm_sq;       // loss
    out[1] = 0.0f;                                  // mean(differences) == 0
    out[2] = (float)(total - (long long)bad);       // good
    out[3] = (float)bad;                            // bad
    out[4] = sqrtf(mean_norm_sq);
  }
}

extern "C" void kernel_launch(void* const* d_in, const int* in_sizes, int n_in,
                              void* d_out, int out_size, void* d_ws, size_t ws_size,
                              hipStream_t stream) {
  (void)in_sizes; (void)n_in; (void)out_size; (void)ws_size;
  const float* h1 = (const float*)d_in[0];
  const float* h2 = (const float*)d_in[1];
  // d_in[2] (h3) is unused by the reference.

  float* G   = (float*)d_ws;                       // 512*512 floats = 1 MB
  float* acc = G + (size_t)TWO_N * TWO_N;          // 4 words of accumulators
  float* out = (float*)d_out;

  zero_acc_kernel     <<<1, 32, 0, stream>>>(acc);
  gram_wmma_tdm_kernel<<<dim3(4, 32), 256, 0, stream>>>(h1, h2, G);
  triplet_stats_kernel<<<TWO_N, HALF_N, 0, stream>>>(G, acc);
  finalize_kernel     <<<1, TWO_N, 0, stream>>>(G, acc, out);
}